// GaussianMRIModel_14087492731379
// MI455X (gfx1250) — compile-verified
//
#include <hip/hip_runtime.h>
#include <hip/hip_bf16.h>

typedef __attribute__((ext_vector_type(2))) float v2f;
typedef __attribute__((ext_vector_type(8))) float v8f;

namespace {
constexpr int   kD = 160, kH = 160, kW = 160;
constexpr int   kDHW  = kD * kH * kW;
constexpr int   kN    = 40000;
constexpr int   kP    = 15;            // patch side (2*7+1)
constexpr float kSpan = 159.0f;        // D-1 == H-1 == W-1
constexpr float kNegHalfLog2e = -0.72134752044448169f;  // -0.5 * log2(e)
}

__global__ void zero_out_kernel(float* __restrict__ out, int n) {
  int i = blockIdx.x * 256 + threadIdx.x;
  if (i < n) out[i] = 0.0f;
}

// One wave32 per Gaussian; per-Gaussian params are block-uniform -> scalar path.
// WMMA tile: rows m = y-offset (0..15), cols n = x-offset (0..15);
// loop over z-offset t (wave-uniform, clipped to the valid/truncated range).
__global__ __launch_bounds__(32) void splat_wmma_kernel(
    const float* __restrict__ centers,
    const float* __restrict__ log_scales,
    const float* __restrict__ quats,
    const float* __restrict__ rho,
    float* __restrict__ out)
{
  const int g    = blockIdx.x;
  const int lane = threadIdx.x;  // 0..31

  // amplitude mask (wave-uniform): sqrt(re^2+im^2) > 1e-6  <=>  re^2+im^2 > 1e-12
  const float rr = rho[2 * g + 0];
  const float ri = rho[2 * g + 1];
  if (rr * rr + ri * ri <= 1e-12f) return;

  const float cz = centers[3 * g + 0] * kSpan;
  const float cy = centers[3 * g + 1] * kSpan;
  const float cx = centers[3 * g + 2] * kSpan;
  const float s0 = expf(log_scales[3 * g + 0]);
  const float s1 = expf(log_scales[3 * g + 1]);
  const float s2 = expf(log_scales[3 * g + 2]);

  float qw = quats[4 * g + 0], qx = quats[4 * g + 1];
  float qy = quats[4 * g + 2], qz = quats[4 * g + 3];
  {
    float n2  = qw * qw + qx * qx + qy * qy + qz * qz;
    float inv = 1.0f / fmaxf(sqrtf(n2), 1e-6f);
    qw *= inv; qx *= inv; qy *= inv; qz *= inv;
  }
  const float ww = qw * qw, xx = qx * qx, yy = qy * qy, zz = qz * qz;
  const float wx = qw * qx, wy = qw * qy, wz = qw * qz;
  const float xy = qx * qy, xz = qx * qz, yz = qy * qz;
  const float R00 = ww + xx - yy - zz, R01 = 2.f * (xy - wz), R02 = 2.f * (xz + wy);
  const float R10 = 2.f * (xy + wz), R11 = ww - xx + yy - zz, R12 = 2.f * (yz - wx);
  const float R20 = 2.f * (xz - wy), R21 = 2.f * (yz + wx), R22 = ww - xx - yy + zz;

  const float c0 = fmaxf(s0, 1e-4f), c1 = fmaxf(s1, 1e-4f), c2 = fmaxf(s2, 1e-4f);
  const float i0 = 1.f / (c0 * c0), i1 = 1.f / (c1 * c1), i2 = 1.f / (c2 * c2);
  // P = R diag(inv_s2) R^T   (index 0<->z, 1<->y, 2<->x)
  const float P00 = R00 * R00 * i0 + R01 * R01 * i1 + R02 * R02 * i2;
  const float P11 = R10 * R10 * i0 + R11 * R11 * i1 + R12 * R12 * i2;
  const float P22 = R20 * R20 * i0 + R21 * R21 * i1 + R22 * R22 * i2;
  const float P01 = R00 * R10 * i0 + R01 * R11 * i1 + R02 * R12 * i2;
  const float P02 = R00 * R20 * i0 + R01 * R21 * i1 + R02 * R22 * i2;
  const float P12 = R10 * R20 * i0 + R11 * R21 * i1 + R12 * R22 * i2;

  // per-dimension patch bounds (lo integer-valued by construction)
  const float rv0 = s0 * 3.f * kSpan, rv1 = s1 * 3.f * kSpan, rv2 = s2 * 3.f * kSpan;
  const int loz = (int)floorf(cz - rv0), hiz = (int)ceilf(cz + rv0);
  const int loy = (int)floorf(cy - rv1), hiy = (int)ceilf(cy + rv1);
  const int lox = (int)floorf(cx - rv2), hix = (int)ceilf(cx + rv2);
  const int oloz = max(0, -loz), ohiz = min(min(kD - 1, hiz) - loz, kP - 1);
  const int oloy = max(0, -loy), ohiy = min(min(kH - 1, hiy) - loy, kP - 1);
  const int olox = max(0, -lox), ohix = min(min(kW - 1, hix) - lox, kP - 1);

  // B operand (4x16, loop invariant): column n -> [1, x_n, x_n^2, 0]^T.
  const int   n  = lane & 15;
  const float xr = (float)(lox + n) - cx;
  v2f bmat;
  bmat.x = (lane < 16) ? 1.0f : xr * xr;
  bmat.y = (lane < 16) ? xr   : 0.0f;
  const bool vx = (n >= olox) && (n <= ohix);

  // Per-lane y-row pieces (rows of A are y-offsets; lanes 16-31 carry K2,K3 consts).
  const float yr = (float)(loy + n) - cy;
  const float uA = 2.f * P01 * yr;       // coeff of zr in A_m
  const float vA = P11 * yr * yr;        // constant part of A_m
  const float wB = 2.f * P12 * yr;       // constant part of B_m

  // y validity bitmask (bits oloy..ohiy), shifted by this lane's D-row offset,
  // combined with x validity -> one bit-test per output value.
  const int moff = (lane >> 4) << 3;     // this lane owns D rows moff..moff+7
  unsigned ymask = 0u;
  if (ohiy >= oloy) ymask = ((1u << (ohiy - oloy + 1)) - 1u) << oloy;
  const unsigned lmask = vx ? (ymask >> moff) : 0u;

  const int laneOff = moff * kW + n;     // per-lane address component

  for (int t = oloz; t <= ohiz; ++t) {   // uniform z loop, fully clipped
    const float zr = (float)(loz + t) - cz;      // wave-uniform
    const float sa = P00 * zr * zr;              // scalar
    const float sb = 2.f * P02 * zr;             // scalar
    const float Am = fmaf(uA, zr, sa) + vA;
    const float Bm = sb + wB;
    v2f amat;
    amat.x = (lane < 16) ? Am : P22;
    amat.y = (lane < 16) ? Bm : 0.0f;

    v8f cacc = {0.f, 0.f, 0.f, 0.f, 0.f, 0.f, 0.f, 0.f};
    // e(m,n) = A_m + B_m * x_n + P22 * x_n^2  (16x16 tile of exponents)
    v8f e = __builtin_amdgcn_wmma_f32_16x16x4_f32(
        false, amat, false, bmat, (short)0, cacc, false, false);

    float* pt = out + (((loz + t) * kH + loy) * kW + lox + laneOff);
#pragma unroll
    for (int r = 0; r < 8; ++r) {
      if ((lmask >> r) & 1u) {
        const float wgt = __builtin_amdgcn_exp2f(e[r] * kNegHalfLog2e);  // raw v_exp_f32
        atomicAdd(pt + r * kW,        wgt * rr);
        atomicAdd(pt + r * kW + kDHW, wgt * ri);
      }
    }
  }
}

extern "C" void kernel_launch(void* const* d_in, const int* in_sizes, int n_in,
                              void* d_out, int out_size, void* d_ws, size_t ws_size,
                              hipStream_t stream) {
  const float* centers    = (const float*)d_in[0];
  const float* log_scales = (const float*)d_in[1];
  const float* quats      = (const float*)d_in[2];
  const float* rho        = (const float*)d_in[3];
  float* out = (float*)d_out;

  const int ntot = 2 * kDHW;
  zero_out_kernel<<<(ntot + 255) / 256, 256, 0, stream>>>(out, ntot);
  splat_wmma_kernel<<<kN, 32, 0, stream>>>(centers, log_scales, quats, rho, out);
}